// SpGraphAttentionLayer_60284160967003
// MI455X (gfx1250) — compile-verified
//
#include <hip/hip_runtime.h>
#include <hip/hip_bf16.h>

#define N_NODES 100000
#define N_EDGES 3200000
#define IN_F 128
#define OUT_F 64

typedef __attribute__((ext_vector_type(2))) float v2f;
typedef __attribute__((ext_vector_type(8))) float v8f;

// ---------------------------------------------------------------------------
// Zero-fill (d_out is poisoned by the harness; rowsum must start at 0)
// ---------------------------------------------------------------------------
__global__ void fill_zero_kernel(float* __restrict__ p, int n) {
    int i = blockIdx.x * blockDim.x + threadIdx.x;
    if (i < n) p[i] = 0.0f;
}

// ---------------------------------------------------------------------------
// h = x @ W using native fp32 WMMA (V_WMMA_F32_16X16X4_F32).
// One wave computes one 16x16 tile of h; 4 waves/block cover a 16x64 row
// block. 100000 = 6250*16 exactly -> no tails, EXEC all-ones (WMMA req).
//
// Fragment layouts per CDNA5 ISA 7.12.2:
//   A (16x4 f32, 2 VGPRs/lane): lanes 0-15 hold rows M=lane, K=0..1;
//                               lanes 16-31 hold rows M=lane-16, K=2..3.
//   B (4x16 f32, 2 VGPRs/lane): lanes 0-15 col N=lane K={0,1};
//                               lanes 16-31 col N=lane-16 K={2,3}.
//   C/D (16x16 f32, 8 VGPRs):   VGPR r: lanes 0-15 -> M=r, N=lane;
//                               lanes 16-31 -> M=8+r, N=lane-16.
// ---------------------------------------------------------------------------
__global__ __launch_bounds__(128) void gemm_wmma_kernel(
    const float* __restrict__ x, const float* __restrict__ W,
    float* __restrict__ h) {
    const int mt   = blockIdx.x;          // 0..6249 (row tile)
    const int nt   = threadIdx.x >> 5;    // 0..3    (col tile = wave id)
    const int lane = threadIdx.x & 31;
    const int half = lane >> 4;           // 0: lanes 0-15, 1: lanes 16-31
    const int lm   = lane & 15;

    const int row  = mt * 16 + lm;        // A row fed by this lane
    const int ncol = nt * 16 + lm;        // B/D column fed by this lane

    const v2f* __restrict__ xrow = (const v2f*)(x + (size_t)row * IN_F);

    v8f c = {};
#pragma unroll
    for (int kt = 0; kt < IN_F / 4; ++kt) {
        // A fragment: x[row][kt*4 + 2*half .. +1] as one float2 load
        v2f a = xrow[kt * 2 + half];
        // B fragment: W[kt*4 + 2*half + {0,1}][ncol]
        const int kb = kt * 4 + half * 2;
        v2f b;
        b.x = W[(kb + 0) * OUT_F + ncol];
        b.y = W[(kb + 1) * OUT_F + ncol];
        // (neg_a, A, neg_b, B, c_mod, C, reuse_a, reuse_b)
        c = __builtin_amdgcn_wmma_f32_16x16x4_f32(
            false, a, false, b, (short)0, c, false, false);
    }

#pragma unroll
    for (int r = 0; r < 8; ++r) {
        const int m = mt * 16 + half * 8 + r;
        h[(size_t)m * OUT_F + ncol] = c[r];
    }
}

// ---------------------------------------------------------------------------
// Per-node attention scores: s1[n] = h[n,:].a1, s2[n] = h[n,:].a2
// One wave per node; float2 per lane covers all 64 features; xor-shuffle
// reduction over the 32 lanes (wave32).
// ---------------------------------------------------------------------------
__global__ __launch_bounds__(256) void score_kernel(
    const float* __restrict__ h, const float* __restrict__ a,
    float* __restrict__ s1, float* __restrict__ s2) {
    const int node = (blockIdx.x * blockDim.x + threadIdx.x) >> 5;
    const int lane = threadIdx.x & 31;
    if (node >= N_NODES) return;

    const v2f hv = ((const v2f*)(h + (size_t)node * OUT_F))[lane];
    const v2f a1 = ((const v2f*)(a))[lane];         // a[0, 0:64]
    const v2f a2 = ((const v2f*)(a + OUT_F))[lane]; // a[0, 64:128]

    float p1 = hv.x * a1.x + hv.y * a1.y;
    float p2 = hv.x * a2.x + hv.y * a2.y;
#pragma unroll
    for (int off = 16; off > 0; off >>= 1) {
        p1 += __shfl_xor(p1, off, 32);
        p2 += __shfl_xor(p2, off, 32);
    }
    if (lane == 0) {
        s1[node] = p1;
        s2[node] = p2;
    }
}

// ---------------------------------------------------------------------------
// Edge aggregation (the roofline-dominant stage; L2-resident):
// one wave per edge. Coalesced 256B gather of h[src] (float2/lane), scaled
// scatter-add into accum[dst] via per-element f32 atomics (L2 atomic units),
// lane 0 accumulates the softmax denominator.
// ---------------------------------------------------------------------------
__global__ __launch_bounds__(256) void edge_kernel(
    const int* __restrict__ dst, const int* __restrict__ src,
    const float* __restrict__ h, const float* __restrict__ s1,
    const float* __restrict__ s2, float* __restrict__ accum,
    float* __restrict__ rowsum) {
    const int e    = (blockIdx.x * blockDim.x + threadIdx.x) >> 5;
    const int lane = threadIdx.x & 31;
    if (e >= N_EDGES) return;

    const int d = dst[e];   // uniform within the wave -> scalarized
    const int s = src[e];

    const float w = expf(-tanhf(s1[d] + s2[s]));

    const v2f hv = ((const v2f*)(h + (size_t)s * OUT_F))[lane];
    float* outp = accum + (size_t)d * OUT_F + lane * 2;
    atomicAdd(outp + 0, w * hv.x);
    atomicAdd(outp + 1, w * hv.y);
    if (lane == 0) atomicAdd(rowsum + d, w);
}

// ---------------------------------------------------------------------------
// out[n][f] /= (rowsum[n] + 1e-8)
// ---------------------------------------------------------------------------
__global__ void norm_kernel(float* __restrict__ out,
                            const float* __restrict__ rowsum) {
    const int i = blockIdx.x * blockDim.x + threadIdx.x;
    if (i < N_NODES * OUT_F) {
        out[i] = out[i] / (rowsum[i >> 6] + 1e-8f);
    }
}

// ---------------------------------------------------------------------------
extern "C" void kernel_launch(void* const* d_in, const int* in_sizes, int n_in,
                              void* d_out, int out_size, void* d_ws, size_t ws_size,
                              hipStream_t stream) {
    const float* x   = (const float*)d_in[0];   // (100000, 128)
    const int*   adj = (const int*)  d_in[1];   // (2, 3200000)
    const float* W   = (const float*)d_in[2];   // (128, 64)
    const float* a   = (const float*)d_in[3];   // (1, 128)
    float* out = (float*)d_out;                 // (100000, 64)

    // Workspace layout (~26.8 MB total)
    char*  ws     = (char*)d_ws;
    float* h      = (float*)ws;                                   // 25.6 MB
    float* s1     = (float*)(ws + (size_t)N_NODES * OUT_F * 4);   // 0.4 MB
    float* s2     = s1 + N_NODES;                                 // 0.4 MB
    float* rowsum = s2 + N_NODES;                                 // 0.4 MB

    const int* dst = adj;            // adj[0]
    const int* src = adj + N_EDGES;  // adj[1]

    // 1. Zero output accumulator and rowsum (re-done every call; replay-safe)
    fill_zero_kernel<<<(N_NODES * OUT_F + 255) / 256, 256, 0, stream>>>(
        out, N_NODES * OUT_F);
    fill_zero_kernel<<<(N_NODES + 255) / 256, 256, 0, stream>>>(
        rowsum, N_NODES);

    // 2. h = x @ W (fp32 WMMA), 6250 blocks x 4 waves
    gemm_wmma_kernel<<<N_NODES / 16, 128, 0, stream>>>(x, W, h);

    // 3. Per-node scores, wave-per-node: 100000 waves / 8 per block
    score_kernel<<<(N_NODES * 32 + 255) / 256, 256, 0, stream>>>(h, a, s1, s2);

    // 4. Edge aggregation, wave-per-edge: 3.2M waves / 8 per block
    edge_kernel<<<(N_EDGES / 8), 256, 0, stream>>>(dst, src, h, s1, s2,
                                                   out, rowsum);

    // 5. Normalize in place
    norm_kernel<<<(N_NODES * OUT_F + 255) / 256, 256, 0, stream>>>(out, rowsum);
}